// MapEmbeddingTower_75531294868021
// MI455X (gfx1250) — compile-verified
//
#include <hip/hip_runtime.h>

typedef __attribute__((ext_vector_type(2))) float v2f;
typedef __attribute__((ext_vector_type(8))) float v8f;

#define NN 10000
#define NE 160000
#define HID 512

// ---------------- degree / norm precompute ----------------
__global__ void k_init_deg(float* deg) {
  int i = blockIdx.x * blockDim.x + threadIdx.x;
  if (i < NN) deg[i] = 1.0f;  // self-loop contributes 1
}

__global__ void k_deg_scatter(const long long* __restrict__ ei, float* deg) {
  int e = blockIdx.x * blockDim.x + threadIdx.x;
  if (e < NE) {
    int dst = (int)ei[NE + e];
    atomicAdd(&deg[dst], 1.0f);
  }
}

__global__ void k_inv(const float* __restrict__ deg, float* __restrict__ inv) {
  int i = blockIdx.x * blockDim.x + threadIdx.x;
  if (i < NN) inv[i] = rsqrtf(deg[i]);  // deg >= 1 always
}

// ---------------- zero accumulator ----------------
__global__ void k_zero(float4* p) {
  int i = blockIdx.x * blockDim.x + threadIdx.x;  // NN*HID/4 elements
  p[i] = make_float4(0.f, 0.f, 0.f, 0.f);
}

// ---------------- fp32 WMMA GEMM: H[M,512] = X[M,K] @ W[K,512] ----------------
// One wave computes one 16x16 tile. Grid covers 625*32 = 20000 tiles, 8 waves/block.
__global__ __launch_bounds__(256) void k_gemm_wmma(const float* __restrict__ A,
                                                   const float* __restrict__ B,
                                                   float* __restrict__ C, int K) {
  const int lane = threadIdx.x & 31;
  const int widx = threadIdx.x >> 5;
  const int w    = blockIdx.x * 8 + widx;
  const int mt   = w >> 5;          // 32 N-tiles per M-tile row
  const int nt   = w & 31;

  const int half  = lane >> 4;      // 0: lanes 0-15, 1: lanes 16-31
  const int l15   = lane & 15;
  const int row   = mt * 16 + l15;  // A row this lane feeds
  const int col   = nt * 16 + l15;  // B column this lane feeds
  const int khalf = half << 1;      // per-lane K sub-offset (0 or 2)

  const float* __restrict__ arow = A + (long)row * K;
  v8f c = {};
  #pragma unroll 8
  for (int k = 0; k < K; k += 4) {
    const int kb = k + khalf;
    v2f a, b;
    a.x = arow[kb];
    a.y = arow[kb + 1];
    b.x = B[(long)kb * HID + col];
    b.y = B[(long)(kb + 1) * HID + col];
    c = __builtin_amdgcn_wmma_f32_16x16x4_f32(
        /*neg_a=*/false, a, /*neg_b=*/false, b,
        /*c_mod=*/(short)0, c, /*reuse_a=*/false, /*reuse_b=*/false);
  }

  // C/D layout: VGPR j -> M = j (+8 for lanes 16-31), N = lane&15
  const int base_m = mt * 16 + (half << 3);
  const int cc     = nt * 16 + l15;
  #pragma unroll
  for (int j = 0; j < 8; ++j)
    C[(long)(base_m + j) * HID + cc] = c[j];
}

// ---------------- edge scatter: agg[dst] += h[src] * inv[src]*inv[dst] ----------------
__global__ __launch_bounds__(128) void k_edge_scatter(const long long* __restrict__ ei,
                                                      const float* __restrict__ h,
                                                      const float* __restrict__ inv,
                                                      float* __restrict__ agg) {
  const int e   = blockIdx.x;
  const int src = (int)ei[e];
  const int dst = (int)ei[NE + e];
  const float nrm = inv[src] * inv[dst];
  const float4* __restrict__ hs = (const float4*)(h + (long)src * HID);
  float* __restrict__ ad = agg + (long)dst * HID;
  const int f = threadIdx.x;         // 0..127, 4 floats each
  float4 v = hs[f];
  atomicAdd(&ad[f * 4 + 0], v.x * nrm);
  atomicAdd(&ad[f * 4 + 1], v.y * nrm);
  atomicAdd(&ad[f * 4 + 2], v.z * nrm);
  atomicAdd(&ad[f * 4 + 3], v.w * nrm);
}

// ---------------- finalize: x = relu(agg + h*inv^2 + bias) ----------------
__global__ void k_finalize(const float4* __restrict__ agg, const float4* __restrict__ h,
                           const float* __restrict__ inv, const float4* __restrict__ bias,
                           float4* __restrict__ out) {
  const int idx = blockIdx.x * blockDim.x + threadIdx.x;  // per float4, NN*HID/4 total
  const int node = idx / (HID / 4);
  const int fc   = idx % (HID / 4);
  const float ii = inv[node];
  const float s  = ii * ii;  // self-loop norm
  float4 a = agg[idx], hh = h[idx], b = bias[fc], r;
  r.x = fmaxf(a.x + hh.x * s + b.x, 0.f);
  r.y = fmaxf(a.y + hh.y * s + b.y, 0.f);
  r.z = fmaxf(a.z + hh.z * s + b.z, 0.f);
  r.w = fmaxf(a.w + hh.w * s + b.w, 0.f);
  out[idx] = r;
}

extern "C" void kernel_launch(void* const* d_in, const int* in_sizes, int n_in,
                              void* d_out, int out_size, void* d_ws, size_t ws_size,
                              hipStream_t stream) {
  (void)in_sizes; (void)n_in; (void)out_size; (void)ws_size;
  const float*     map = (const float*)d_in[0];
  const long long* ei  = (const long long*)d_in[1];   // int64 [2, NE]
  const float* W1 = (const float*)d_in[2];
  const float* b1 = (const float*)d_in[3];
  const float* Wm = (const float*)d_in[4];            // [4, 512, 512]
  const float* bm = (const float*)d_in[5];            // [4, 512]
  const float* Wl = (const float*)d_in[6];
  const float* bl = (const float*)d_in[7];
  float* out = (float*)d_out;

  char*  ws   = (char*)d_ws;
  float* inv  = (float*)ws;                       // 10000 floats (64KB slot)
  float* h    = (float*)(ws + 65536);             // 10000*512
  float* agg  = h   + (long)NN * HID;             // 10000*512
  float* xbuf = agg + (long)NN * HID;             // 10000*512
  float* deg  = agg;                              // temp reuse before first layer

  const int VEC4 = NN * HID / 4;                  // 1,280,000

  // normalization coefficients
  k_init_deg   <<<(NN + 255) / 256, 256, 0, stream>>>(deg);
  k_deg_scatter<<<(NE + 255) / 256, 256, 0, stream>>>(ei, deg);
  k_inv        <<<(NN + 255) / 256, 256, 0, stream>>>(deg, inv);

  struct Layer { const float* x; int K; const float* W; const float* b; float* y; };
  Layer layers[6] = {
    { map,  128, W1,                 b1,       xbuf },
    { xbuf, 512, Wm + 0 * 512 * 512, bm + 0*512, xbuf },
    { xbuf, 512, Wm + 1 * 512 * 512, bm + 1*512, xbuf },
    { xbuf, 512, Wm + 2 * 512 * 512, bm + 2*512, xbuf },
    { xbuf, 512, Wm + 3 * 512 * 512, bm + 3*512, xbuf },
    { xbuf, 512, Wl,                 bl,       out  },
  };

  for (int l = 0; l < 6; ++l) {
    k_zero        <<<VEC4 / 256, 256, 0, stream>>>((float4*)agg);
    k_gemm_wmma   <<<2500, 256, 0, stream>>>(layers[l].x, layers[l].W, h, layers[l].K);
    k_edge_scatter<<<NE, 128, 0, stream>>>(ei, h, inv, agg);
    k_finalize    <<<VEC4 / 256, 256, 0, stream>>>((const float4*)agg, (const float4*)h,
                                                   inv, (const float4*)layers[l].b,
                                                   (float4*)layers[l].y);
  }
}